// ProgressiveLoss_8830452760987
// MI455X (gfx1250) — compile-verified
//
#include <hip/hip_runtime.h>

#define BATCH 16
#define NBOX  32
#define HH    192
#define WW    192
#define HW    (HH*WW)      // 36864
#define NCH   69
#define NCLS  64
#define NCAND 397          // 1 + 4*99 spiral candidates
#define INV_PI2_4 0.405284734569351f   // 4/pi^2

typedef __attribute__((ext_vector_type(2))) float v2f;
typedef __attribute__((ext_vector_type(8))) float v8f;

__device__ __forceinline__ float softplusf(float z) {
  // jax.nn.softplus: max(z,0) + log1p(exp(-|z|))
  return fmaxf(z, 0.f) + log1pf(expf(-fabsf(z)));
}
__device__ __forceinline__ float wredsum(float v) {
#pragma unroll
  for (int m = 16; m; m >>= 1) v += __shfl_xor(v, m, 32);
  return v;
}

// ---------------------------------------------------------------- init
__global__ void k_init(float* p1, float* acc) {
  int i = threadIdx.x;
  if (i < BATCH) p1[i] = 0.f;
  if (i < BATCH * 10) acc[i] = 0.f;
}

// ------------------------------------------------- per-image serial part
// 1 wave per image: mask build, stable sort, sequential spiral scan,
// gathered-cell CIoU + class losses. Mask lives in LDS (36KB).
__global__ __launch_bounds__(32) void k_sortscan(
    const float* __restrict__ x, const float* __restrict__ bboxes,
    const int* __restrict__ labels, const float* __restrict__ sm,
    unsigned int* __restrict__ mask_ws, float* __restrict__ sb_ws,
    int* __restrict__ sl_ws, float* __restrict__ p1_ws)
{
  __shared__ unsigned int maskw[HW / 4];
  __shared__ float sbl[NBOX][4];
  __shared__ int   sll[NBOX];
  __shared__ int   ordl[NBOX];
  __shared__ int   cysl[NBOX], cxsl[NBOX];

  const int img  = blockIdx.x;
  const int lane = threadIdx.x;
  const float* xb = x + (size_t)img * NCH * HW;
  unsigned char* maskb = (unsigned char*)maskw;

  // mask0: sigmoid(z) > 0.5  <=>  z > 0
  for (int i = lane; i < HW; i += 32) maskb[i] = (xb[i] > 0.f) ? 1 : 0;

  // stable argsort of 32 boxes by |area|, ascending (rank via shuffles)
  const float* bx = bboxes + (size_t)img * NBOX * 4;
  float bl0 = bx[lane*4+0], bt0 = bx[lane*4+1], br0 = bx[lane*4+2], bb0 = bx[lane*4+3];
  float area = fabsf((br0 - bl0) * (bb0 - bt0));
  int rank = 0;
  for (int j = 0; j < NBOX; ++j) {
    float aj = __shfl(area, j, 32);
    if (aj < area || (aj == area && j < lane)) ++rank;
  }
  ordl[rank] = lane;
  __syncthreads();
  int oi = ordl[lane];
  sbl[lane][0] = bx[oi*4+0]; sbl[lane][1] = bx[oi*4+1];
  sbl[lane][2] = bx[oi*4+2]; sbl[lane][3] = bx[oi*4+3];
  sll[lane] = labels[img*NBOX + oi];
  __syncthreads();

  // sequential scan: spiral cell search per sorted box, mark mask=2
  for (int k = 0; k < NBOX; ++k) {
    float gl = sbl[k][0], gt = sbl[k][1], gr = sbl[k][2], gb = sbl[k][3];
    int cx = (int)rintf((gl + gr) * 0.5f * 0.25f);   // round-half-even
    int cy = (int)rintf((gt + gb) * 0.5f * 0.25f);
    if (cy >= HH) cy -= 1;
    if (cx >= WW) cx -= 1;
    float cxf = (float)cx, cyf = (float)cy;
    float me = fmaxf(gr - gl, gb - gt);
    float dx = (gr - gl) / me, dy = (gb - gt) / me;

    int best = NCAND;
    for (int idx = lane; idx < NCAND; idx += 32) {
      float xf, yf;
      if (idx == 0) { xf = cxf; yf = cyf; }
      else {
        int i = idx - 1; float fs = (float)((i >> 2) + 1); int q = i & 3;
        xf = cxf + ((q & 1) ? fs * dx : -fs * dx);
        yf = cyf + ((q & 2) ? fs * dy : -fs * dy);
      }
      int xi = (int)rintf(xf), yi = (int)rintf(yf);
      bool inb = (xi >= 0) && (xi < WW) && (yi >= 0) && (yi < HH);
      int xc = min(max(xi, 0), WW - 1), yc = min(max(yi, 0), HH - 1);
      unsigned char mv = maskb[yc * WW + xc];
      if (inb && mv != 2) best = min(best, idx);
    }
#pragma unroll
    for (int m = 16; m; m >>= 1) best = min(best, __shfl_xor(best, m, 32));

    int cyk, cxk;
    if (best < NCAND) {
      float xf, yf;
      if (best == 0) { xf = cxf; yf = cyf; }
      else {
        int i = best - 1; float fs = (float)((i >> 2) + 1); int q = i & 3;
        xf = cxf + ((q & 1) ? fs * dx : -fs * dx);
        yf = cyf + ((q & 2) ? fs * dy : -fs * dy);
      }
      cxk = (int)rintf(xf); cyk = (int)rintf(yf);
    } else { cyk = HH - 1; cxk = WW - 1; }
    if (lane == 0) { maskb[cyk * WW + cxk] = 2; cysl[k] = cyk; cxsl[k] = cxk; }
    __syncthreads();
  }

  // gathered-cell losses: lane k handles sorted box k
  int cell = cysl[lane] * WW + cxsl[lane];
  float pl = xb[1*HW+cell], pt = xb[2*HW+cell], pr = xb[3*HW+cell], pb = xb[4*HW+cell];
  float gl = sbl[lane][0], gt = sbl[lane][1], gr = sbl[lane][2], gb = sbl[lane][3];
  const float eps = 1e-10f;
  float w1 = pr - pl, h1 = pb - pt, w2 = gr - gl, h2 = gb - gt;
  float iw = fmaxf(fminf(pr, gr) - fmaxf(pl, gl), 0.f);
  float ih = fmaxf(fminf(pb, gb) - fmaxf(pt, gt), 0.f);
  float inter = iw * ih;
  float uni = w1*h1 + w2*h2 - inter + eps;
  float iou = inter / uni;
  float cw = fmaxf(pr, gr) - fminf(pl, gl);
  float chh = fmaxf(pb, gb) - fminf(pt, gt);
  float c2 = cw*cw + chh*chh + eps;
  float dxx = gl + gr - pl - pr, dyy = gb + gt - pb - pt;
  float rho2 = (dxx*dxx + dyy*dyy) * 0.25f;
  float dat = atanf(w2 / (h2 + eps)) - atanf(w1 / (h1 + eps));
  float v = INV_PI2_4 * dat * dat;
  float alpha = v / (v - iou + (1.f + eps));
  float ciou = iou - (rho2 / c2 + v * alpha);
  float bbox1 = -wredsum(ciou) / (float)NBOX;

  int lab = sll[lane];
  float dot = 0.f;
  for (int f = 0; f < NCLS; ++f) dot += xb[(size_t)(5 + f) * HW + cell] * sm[lab * NCLS + f];
  float cls1 = -wredsum(dot) / (float)NBOX;

  if (lane == 0) p1_ws[img] = bbox1 + cls1;

  // export sorted boxes/labels + final mask for the big pass
  sb_ws[(img*NBOX + lane)*4 + 0] = gl; sb_ws[(img*NBOX + lane)*4 + 1] = gt;
  sb_ws[(img*NBOX + lane)*4 + 2] = gr; sb_ws[(img*NBOX + lane)*4 + 3] = gb;
  sl_ws[img*NBOX + lane] = lab;
  unsigned int* mw = mask_ws + (size_t)img * (HW / 4);
  for (int i = lane; i < HW / 4; i += 32) mw[i] = maskw[i];
}

// ---------------------------------------------- per-cell heavy pass
// grid (144, 16), 256 threads: each thread owns one cell; each wave owns 32
// cells = two 16-row WMMA tiles. sim = learnt[:,5:69] @ SM^T via f32 WMMA.
__global__ __launch_bounds__(256) void k_cells(
    const float* __restrict__ x, const float* __restrict__ sm,
    const unsigned int* __restrict__ mask_ws, const float* __restrict__ sb_ws,
    const int* __restrict__ sl_ws, float* __restrict__ acc_ws)
{
  __shared__ float smT[NCLS * NCLS];   // [feature][class] (conflict-free B reads)
  __shared__ float binfo[NBOX][8];     // l,t,r,b,area,xc,yc,atan(w/h)
  __shared__ int   sll[NBOX];
  __shared__ float simbuf[8][256];     // per-wave 16x16 D-tile staging

  const int img  = blockIdx.y;
  const int tid  = threadIdx.x;
  const int lane = tid & 31;
  const int wave = tid >> 5;

  for (int i = tid; i < NCLS * NCLS; i += 256) {
    int c = i >> 6, f = i & 63;
    smT[f * NCLS + c] = sm[i];
  }
  if (tid < NBOX) {
    const float* sb = sb_ws + (img * NBOX + tid) * 4;
    float l = sb[0], t = sb[1], r = sb[2], bt = sb[3];
    binfo[tid][0] = l; binfo[tid][1] = t; binfo[tid][2] = r; binfo[tid][3] = bt;
    binfo[tid][4] = (r - l) * (bt - t);
    binfo[tid][5] = (l + r) * 0.5f; binfo[tid][6] = (t + bt) * 0.5f;
    binfo[tid][7] = atanf((r - l) / ((bt - t) + 1e-7f));
    sll[tid] = sl_ws[img * NBOX + tid];
  }
  __syncthreads();

  const float* xb = x + (size_t)img * NCH * HW;
  const int cbase = blockIdx.x * 256 + wave * 32;
  const int cell  = cbase + lane;

  float z  = xb[cell];
  float l1 = xb[1*HW+cell], t1 = xb[2*HW+cell], r1 = xb[3*HW+cell], b1 = xb[4*HW+cell];
  const unsigned char* maskb = (const unsigned char*)(mask_ws + (size_t)img * (HW / 4));
  int mv = maskb[cell];

  // pairwise CIoU vs 32 sorted boxes: max + first-argmax
  const float eps = 1e-7f;
  float wp = r1 - l1, hp = b1 - t1;
  float area1 = wp * hp;
  float xp = (l1 + r1) * 0.5f, yp = (t1 + b1) * 0.5f;
  float atanp = atanf(wp / (hp + eps));
  float bestv = -INFINITY; int besti = 0;
  for (int j = 0; j < NBOX; ++j) {
    float l2 = binfo[j][0], t2 = binfo[j][1], r2 = binfo[j][2], b2 = binfo[j][3];
    float iw = fmaxf(fminf(r1, r2) - fmaxf(l1, l2), 0.f);
    float ih = fmaxf(fminf(b1, b2) - fmaxf(t1, t2), 0.f);
    float inter = iw * ih;
    float uni = area1 + binfo[j][4] - inter + eps;
    float iou = inter / uni;
    float wi = fmaxf(fmaxf(r1, r2) - fminf(l1, l2), 0.f);
    float hi = fmaxf(fmaxf(b1, b2) - fminf(t1, t2), 0.f);
    float diag2 = wi*wi + hi*hi + eps;
    float dxc = xp - binfo[j][5], dyc = yp - binfo[j][6];
    float cent2 = dxc*dxc + dyc*dyc;
    float dat = binfo[j][7] - atanp;
    float vv = INV_PI2_4 * dat * dat;
    float alpha = vv / (1.f - iou + vv + eps);
    float ciou = iou - cent2 / diag2 - alpha * vv;
    if (ciou > bestv) { bestv = ciou; besti = j; }
  }

  // sim[cell, 0:64] via V_WMMA_F32_16X16X4_F32 (K=64 -> 16 k-steps, 4 n-tiles)
  float clsv = -INFINITY; int clsi = 0;
  const int koff = (lane >> 4) * 2;   // A/B lane K offset per ISA 16x4 layout
  const int mrow = lane & 15;
#pragma unroll
  for (int t = 0; t < 2; ++t) {       // two 16-cell M-tiles per wave
    const int mcell = cbase + t * 16 + mrow;
#pragma unroll
    for (int n = 0; n < 4; ++n) {     // 4 class tiles of 16
      v8f acc = {};
#pragma unroll
      for (int k = 0; k < 16; ++k) {
        int f = 4 * k + koff;
        v2f av, bv;
        av.x = xb[(size_t)(5 + f) * HW + mcell];
        av.y = xb[(size_t)(6 + f) * HW + mcell];
        bv.x = smT[f * NCLS + 16 * n + mrow];
        bv.y = smT[(f + 1) * NCLS + 16 * n + mrow];
        acc = __builtin_amdgcn_wmma_f32_16x16x4_f32(false, av, false, bv,
                                                    (short)0, acc, false, false);
      }
      // spill D tile to LDS so each lane can read its own cell's 16 classes
#pragma unroll
      for (int r = 0; r < 8; ++r)
        simbuf[wave][(r + ((lane >> 4) << 3)) * 16 + mrow] = acc[r];
      asm volatile("s_wait_dscnt 0" ::: "memory");
      if ((lane >> 4) == t) {
#pragma unroll
        for (int j = 0; j < 16; ++j) {
          float sv = simbuf[wave][mrow * 16 + j];
          if (sv > clsv) { clsv = sv; clsi = 16 * n + j; }
        }
      }
      asm volatile("s_wait_dscnt 0" ::: "memory");
    }
  }

  // masked statistics
  bool sel1 = (mv == 1);
  bool neg  = sel1 && (bestv <  0.f);
  bool pos  = sel1 && (bestv >= 0.f);
  bool mm   = pos && (sll[besti] == clsi);
  float sp  = softplusf(z);
  float q[10];
  q[0] = (mv == 2) ? (sp - z) : 0.f;  q[1] = (mv == 2) ? 1.f : 0.f;  // bce(conf,1,sel2)
  q[2] = (mv == 0) ? sp : 0.f;        q[3] = (mv == 0) ? 1.f : 0.f;  // bce(conf,0,sel0)
  q[4] = neg ? sp : 0.f;              q[5] = neg ? 1.f : 0.f;        // bce(conf,0,neg)
  q[6] = mm ? (sp - 0.9f * z) : 0.f;  q[7] = mm ? 1.f : 0.f;         // bce(conf,0.9,m)
  q[8] = mm ? bestv : 0.f;                                           // sum(values*m)
  q[9] = mm ? clsv  : 0.f;                                           // sum(cls_values*m)
#pragma unroll
  for (int i = 0; i < 10; ++i) {
    float rs = wredsum(q[i]);
    if (lane == 0) atomicAdd(&acc_ws[img * 10 + i], rs);
  }
}

// ---------------------------------------------------------------- finalize
__global__ void k_final(const float* __restrict__ p1, const float* __restrict__ acc,
                        float* __restrict__ out) {
  int lane = threadIdx.x;
  float L = 0.f;
  if (lane < BATCH) {
    const float* a = acc + lane * 10;
    L = p1[lane];
    L += (a[1] > 0.f) ? a[0] / fmaxf(a[1], 1.f) : 0.f;
    L += (a[3] > 0.f) ? a[2] / fmaxf(a[3], 1.f) : 0.f;
    L += (a[5] > 0.f) ? a[4] / fmaxf(a[5], 1.f) : 0.f;
    L += (a[7] > 0.f) ? a[6] / fmaxf(a[7], 1.f) : 0.f;
    L += (a[7] > 0.f) ? (-(a[8] + a[9]) / fmaxf(a[7], 1.f)) : 0.f;
  }
  L = wredsum(L);
  if (lane == 0) out[0] = L;
}

extern "C" void kernel_launch(void* const* d_in, const int* in_sizes, int n_in,
                              void* d_out, int out_size, void* d_ws, size_t ws_size,
                              hipStream_t stream) {
  const float* x      = (const float*)d_in[0];
  const float* bboxes = (const float*)d_in[1];
  const int*   labels = (const int*)d_in[2];
  const float* sm     = (const float*)d_in[3];

  char* ws = (char*)d_ws;
  unsigned int* mask_ws = (unsigned int*)ws;                     // 589824 B
  float* sb_ws  = (float*)(ws + 589824);                         //   8192 B
  int*   sl_ws  = (int*)  (ws + 589824 + 8192);                  //   2048 B
  float* p1_ws  = (float*)(ws + 589824 + 8192 + 2048);           //     64 B
  float* acc_ws = (float*)(ws + 589824 + 8192 + 2048 + 64);      //    640 B

  k_init<<<1, 256, 0, stream>>>(p1_ws, acc_ws);
  k_sortscan<<<BATCH, 32, 0, stream>>>(x, bboxes, labels, sm,
                                       mask_ws, sb_ws, sl_ws, p1_ws);
  k_cells<<<dim3(HW / 256, BATCH), 256, 0, stream>>>(x, sm, mask_ws,
                                                     sb_ws, sl_ws, acc_ws);
  k_final<<<1, 32, 0, stream>>>(p1_ws, acc_ws, (float*)d_out);
}